// GraphSAGELayer_42288247996604
// MI455X (gfx1250) — compile-verified
//
#include <hip/hip_runtime.h>
#include <hip/hip_bf16.h>

#define NNODES 100000
#define NEDGES 1600000
#define D 128      // IN_FEATS == OUT_FEATS
#define KDIM 256   // 2*D
#define LNEPS 1e-5f

typedef __attribute__((ext_vector_type(2))) float v2f;
typedef __attribute__((ext_vector_type(8))) float v8f;

// ---------------------------------------------------------------- zero scratch
__global__ __launch_bounds__(256) void zero_ws_kernel(float* __restrict__ ws, int n4) {
    int i = blockIdx.x * 256 + threadIdx.x;
    if (i < n4) reinterpret_cast<float4*>(ws)[i] = make_float4(0.f, 0.f, 0.f, 0.f);
}

// ------------------------------------------------- edge-parallel scatter-add
// 32 consecutive threads handle one edge (128 floats, 4 per thread).
// h and ah are ~51MB each -> L2-resident on MI455X (192MB L2); the
// global_atomic_add_f32 RMWs execute at the L2 atomic units.
__global__ __launch_bounds__(256) void edge_aggregate_kernel(
    const float* __restrict__ h, const int* __restrict__ src,
    const int* __restrict__ dst, float* __restrict__ deg,
    float* __restrict__ ah) {
    long long gid = (long long)blockIdx.x * 256 + threadIdx.x;
    int e   = (int)(gid >> 5);
    int sub = (int)(gid & 31);
    if (e >= NEDGES) return;
    int s = src[e], d = dst[e];
    const float4 v = reinterpret_cast<const float4*>(h + (size_t)s * D)[sub];
    float* o = ah + (size_t)d * D + sub * 4;
    atomicAdd(o + 0, v.x);
    atomicAdd(o + 1, v.y);
    atomicAdd(o + 2, v.z);
    atomicAdd(o + 3, v.w);
    if (sub == 0) atomicAdd(deg + d, 1.0f);
}

// ------------------------------------------------------------ async LDS stage
__device__ __forceinline__ void async_copy_b128(const float* gptr, const float* lptr) {
    unsigned lds = (unsigned)(size_t)lptr;                  // low 32 bits = LDS addr
    unsigned long long ga = (unsigned long long)(size_t)gptr;
    asm volatile("global_load_async_to_lds_b128 %0, %1, off"
                 :: "v"(lds), "v"(ga) : "memory");
}
__device__ __forceinline__ void wait_async0() {
    asm volatile("s_wait_asynccnt 0x0" ::: "memory");
}

// -------------------------------- fused [h|ah/deg] @ W^T + b, LayerNorm, ReLU
// Block = 128 threads (4 waves), 32 nodes x 128 outputs.
// Each wave: 2x2 grid of 16x16 tiles, split into h-half / ah-half accumulator
// sets so that all global->LDS staging is a raw copy (async b128 TDM path);
// the 1/deg norm is applied once at accumulator writeback:
//   y = h @ W1^T + diag(nrm) * (ah @ W2^T)
#define CAT_LD 260   // 1040B rows (16B aligned); bank stride 4 -> conflict-free
#define W_LD   36    //  144B rows (16B aligned); 36*lm mod 64 all-distinct
#define Y_LD   132
#define KCW    32    // K-chunk width staged in LDS

#define CAT_FLOATS (32 * CAT_LD)      // 8320
#define W_FLOATS   (128 * W_LD)       // 4608 ( >= 32*Y_LD = 4224, aliased by y )

__global__ __launch_bounds__(128) void gemm_ln_relu_kernel(
    const float* __restrict__ h, const float* __restrict__ ah,
    const float* __restrict__ deg, const float* __restrict__ W,
    const float* __restrict__ bias, const float* __restrict__ gamma,
    const float* __restrict__ beta, float* __restrict__ out) {
    __shared__ float smem[CAT_FLOATS + W_FLOATS];
    __shared__ float nrm[32];
    float (*cat)[CAT_LD] = (float(*)[CAT_LD])smem;
    float (*wl)[W_LD]    = (float(*)[W_LD])(smem + CAT_FLOATS);
    float (*yl)[Y_LD]    = (float(*)[Y_LD])(smem + CAT_FLOATS);  // aliases wl

    const int t    = threadIdx.x;
    const int base = blockIdx.x * 32;

    if (t < 32) {
        float dg = deg[base + t];
        nrm[t] = dg > 0.f ? 1.f / dg : 0.f;
    }

    // async-stage cat = [h | ah] : 32 x 256, raw 16B copies
#pragma unroll
    for (int j = 0; j < 8; ++j) {
        int i4 = t + 128 * j;                 // 1024 float4 per half
        int m = i4 >> 5, c = (i4 & 31) * 4;
        async_copy_b128(h  + (size_t)(base + m) * D + c, &cat[m][c]);
        async_copy_b128(ah + (size_t)(base + m) * D + c, &cat[m][128 + c]);
    }
    wait_async0();
    __syncthreads();

    const int lane = t & 31;
    const int wave = t >> 5;        // 0..3 : owns cols [32*wave, 32*wave+32)
    const int half = lane >> 4;     // 0: K={k,k+1}, 1: K={k+2,k+3}
    const int lm   = lane & 15;
    const int c0   = wave * 32 + lm;
    const int c1   = wave * 32 + 16 + lm;

    v8f aH00 = {}, aH01 = {}, aH10 = {}, aH11 = {};
    v8f aA00 = {}, aA01 = {}, aA10 = {}, aA11 = {};

    auto run_half = [&](int kcBeg, v8f& A00, v8f& A01, v8f& A10, v8f& A11) {
        for (int kc = kcBeg; kc < kcBeg + 128; kc += KCW) {
            __syncthreads();
            // async-stage 128 x 32 chunk of W (row-major [out][2*in])
#pragma unroll
            for (int j = 0; j < 8; ++j) {
                int i4 = t + 128 * j;
                int n = i4 >> 3, kk = (i4 & 7) * 4;
                async_copy_b128(W + (size_t)n * KDIM + kc + kk, &wl[n][kk]);
            }
            wait_async0();
            __syncthreads();
#pragma unroll
            for (int k = 0; k < KCW; k += 4) {
                const int ka = kc + k + 2 * half;   // global K for A (cat)
                const int kb = k + 2 * half;        // local K for B (wl)
                v2f a0, a1, b0, b1;
                a0.x = cat[lm][ka];        a0.y = cat[lm][ka + 1];
                a1.x = cat[16 + lm][ka];   a1.y = cat[16 + lm][ka + 1];
                b0.x = wl[c0][kb];         b0.y = wl[c0][kb + 1];
                b1.x = wl[c1][kb];         b1.y = wl[c1][kb + 1];
                A00 = __builtin_amdgcn_wmma_f32_16x16x4_f32(
                    false, a0, false, b0, (short)0, A00, false, false);
                A01 = __builtin_amdgcn_wmma_f32_16x16x4_f32(
                    false, a0, false, b1, (short)0, A01, false, false);
                A10 = __builtin_amdgcn_wmma_f32_16x16x4_f32(
                    false, a1, false, b0, (short)0, A10, false, false);
                A11 = __builtin_amdgcn_wmma_f32_16x16x4_f32(
                    false, a1, false, b1, (short)0, A11, false, false);
            }
        }
    };
    run_half(0,   aH00, aH01, aH10, aH11);    // h-half of K
    run_half(128, aA00, aA01, aA10, aA11);    // ah-half of K

    __syncthreads();   // all waves done reading wl before yl aliases it
    // C layout: VGPR r -> row r + 8*half (+16 for row tile 1), column c0/c1
#pragma unroll
    for (int r = 0; r < 8; ++r) {
        int row = r + 8 * half;
        float n0 = nrm[row], n1 = nrm[row + 16];
        yl[row][c0]      = aH00[r] + n0 * aA00[r];
        yl[row][c1]      = aH01[r] + n0 * aA01[r];
        yl[row + 16][c0] = aH10[r] + n1 * aA10[r];
        yl[row + 16][c1] = aH11[r] + n1 * aA11[r];
    }
    __syncthreads();

    // LayerNorm + affine + ReLU: 4 threads per node row, 32 cols each
    const int r = t >> 2;          // 0..31
    const int g = t & 3;
    float vals[32];
    float s = 0.f;
#pragma unroll
    for (int j = 0; j < 32; ++j) {
        int c = g + 4 * j;
        float v = yl[r][c] + bias[c];
        vals[j] = v;
        s += v;
    }
    s += __shfl_xor(s, 1, 32);
    s += __shfl_xor(s, 2, 32);
    float mean = s * (1.f / 128.f);
    float s2 = 0.f;
#pragma unroll
    for (int j = 0; j < 32; ++j) { float dv = vals[j] - mean; s2 += dv * dv; }
    s2 += __shfl_xor(s2, 1, 32);
    s2 += __shfl_xor(s2, 2, 32);
    float inv = rsqrtf(s2 * (1.f / 128.f) + LNEPS);
    const int node = base + r;
#pragma unroll
    for (int j = 0; j < 32; ++j) {
        int c = g + 4 * j;
        float o = (vals[j] - mean) * inv * gamma[c] + beta[c];
        out[(size_t)node * D + c] = fmaxf(o, 0.f);
    }
}

// ---------------------------------------------------------------------- launch
extern "C" void kernel_launch(void* const* d_in, const int* in_sizes, int n_in,
                              void* d_out, int out_size, void* d_ws, size_t ws_size,
                              hipStream_t stream) {
    const float* h     = (const float*)d_in[0];
    const float* W     = (const float*)d_in[1];
    const float* b     = (const float*)d_in[2];
    const float* gamma = (const float*)d_in[3];
    const float* beta  = (const float*)d_in[4];
    const int*   src   = (const int*)d_in[5];
    const int*   dst   = (const int*)d_in[6];
    float*       out   = (float*)d_out;

    float* ws = (float*)d_ws;
    const int degPad = (NNODES + 255) & ~255;   // 100096 floats
    float* deg = ws;
    float* ah  = ws + degPad;

    // zero deg + ah every call (atomics accumulate)
    int n4 = (degPad + NNODES * D) / 4;
    zero_ws_kernel<<<(n4 + 255) / 256, 256, 0, stream>>>(ws, n4);

    long long ethreads = (long long)NEDGES * 32;
    int eblocks = (int)((ethreads + 255) / 256);
    edge_aggregate_kernel<<<eblocks, 256, 0, stream>>>(h, src, dst, deg, ah);

    int nblocks = (NNODES + 31) / 32;           // 3125, exact
    gemm_ln_relu_kernel<<<nblocks, 128, 0, stream>>>(h, ah, deg, W, b, gamma,
                                                     beta, out);
}